// DGCNN_semseg_37151467111298
// MI455X (gfx1250) — compile-verified
//
#include <hip/hip_runtime.h>
#include <math.h>

#define Bb 2
#define Nn 4096
#define Kk 20
#define BN  (Bb * Nn)          // 8192
#define BNK (BN * Kk)          // 163840
#define EPSV 1e-5f

typedef __attribute__((ext_vector_type(2))) float v2f;
typedef __attribute__((ext_vector_type(8))) float v8f;

// ---------------------------------------------------------------------------
// 0. transpose input (B,N,4) -> channel-major [4][B*N]
__global__ void k_transpose_in(const float* __restrict__ x, float* __restrict__ xc) {
    int p = blockIdx.x * blockDim.x + threadIdx.x;
    if (p >= BN) return;
#pragma unroll
    for (int c = 0; c < 4; ++c) xc[c * BN + p] = x[p * 4 + c];
}

// ---------------------------------------------------------------------------
// 1. kNN over first 3 channels of f[C][BN]; per-batch candidates; top-20 by
//    d = 2*dot - |p|^2 - |q|^2 (stable insertion matches lax.top_k ordering).
__global__ void k_knn(const float* __restrict__ f, int* __restrict__ idx) {
    __shared__ float sx[128], sy[128], sz[128];
    int p = blockIdx.x * 128 + threadIdx.x;     // blockDim = 128, N % 128 == 0
    int b = p / Nn;
    int base = b * Nn;
    float px = f[0 * BN + p], py = f[1 * BN + p], pz = f[2 * BN + p];
    float pp = px * px + py * py + pz * pz;
    float best[Kk]; int bidx[Kk];
#pragma unroll
    for (int i = 0; i < Kk; ++i) { best[i] = -3.4e38f; bidx[i] = 0; }
    for (int m0 = 0; m0 < Nn; m0 += 128) {
        __syncthreads();
        int ml = base + m0 + threadIdx.x;
        sx[threadIdx.x] = f[0 * BN + ml];
        sy[threadIdx.x] = f[1 * BN + ml];
        sz[threadIdx.x] = f[2 * BN + ml];
        __syncthreads();
        for (int t = 0; t < 128; ++t) {
            float qx = sx[t], qy = sy[t], qz = sz[t];
            float qq = qx * qx + qy * qy + qz * qz;
            float d  = 2.f * (px * qx + py * qy + pz * qz) - pp - qq;
            if (d > best[Kk - 1]) {
                int j = Kk - 1;
                while (j > 0 && d > best[j - 1]) {
                    best[j] = best[j - 1]; bidx[j] = bidx[j - 1]; --j;
                }
                best[j] = d; bidx[j] = m0 + t;
            }
        }
    }
    for (int i = 0; i < Kk; ++i) idx[p * Kk + i] = bidx[i];
}

// ---------------------------------------------------------------------------
// 2. edge features: out[2C][BN*K]; first C = nbr-ctr, next C = ctr
__global__ void k_edge(const float* __restrict__ f, const int* __restrict__ idx,
                       float* __restrict__ out, int C) {
    long t = (long)blockIdx.x * blockDim.x + threadIdx.x;  // over BN*K
    if (t >= (long)BNK) return;
    int p = (int)(t / Kk);
    int b = p / Nn;
    long q = (long)b * Nn + idx[t];
    for (int c = 0; c < C; ++c) {
        float ctr = f[(long)c * BN + p];
        float nb  = f[(long)c * BN + q];
        out[(long)c * BNK + t]        = nb - ctr;
        out[(long)(C + c) * BNK + t]  = ctr;
    }
}

// ---------------------------------------------------------------------------
// 3a. FAST WMMA fp32 GEMM: Y[M][P] = W[M][Kc] @ X[Kc][P] (+bias/mode for head)
//     Requires: M % 16 == 0, Kc % 4 == 0, P % 64 == 0. No guards in hot loop;
//     EXEC stays all-ones (WMMA requirement). One wave owns a 16x64 tile with
//     4 accumulators so a single A fetch feeds 4 WMMAs per K-step.
__global__ void k_gemm_fast(const float* __restrict__ W, const float* __restrict__ X,
                            float* __restrict__ Y, int M, int Kc, long P,
                            const float* __restrict__ bias, int mode) {
    const int wave = threadIdx.x >> 5;
    const int lane = threadIdx.x & 31;
    const long pTiles = P >> 6;                              // 64-wide tiles
    long tile = (long)blockIdx.x * (blockDim.x >> 5) + wave; // wave-uniform
    if (tile >= (long)(M >> 4) * pTiles) return;             // whole-wave exit
    const int  tm = (int)(tile / pTiles) << 4;
    const long tn = (tile % pTiles) << 6;
    const int half  = lane >> 4;                             // 0 | 1
    const int l16   = lane & 15;
    const int kbase = half * 2;

    const float* Wrow = W + (long)(tm + l16) * Kc + kbase;   // a.x @ +k0, a.y @ +k0+1
    const float* Xcol = X + tn + l16;

    v8f acc0 = {}, acc1 = {}, acc2 = {}, acc3 = {};
#pragma unroll 2
    for (int k0 = 0; k0 < Kc; k0 += 4) {
        v2f a;
        a.x = Wrow[k0];
        a.y = Wrow[k0 + 1];
        const float* xr0 = Xcol + (long)(k0 + kbase) * P;
        const float* xr1 = xr0 + P;
        v2f b0, b1, b2, b3;
        b0.x = xr0[0];  b0.y = xr1[0];
        b1.x = xr0[16]; b1.y = xr1[16];
        b2.x = xr0[32]; b2.y = xr1[32];
        b3.x = xr0[48]; b3.y = xr1[48];
        acc0 = __builtin_amdgcn_wmma_f32_16x16x4_f32(false, a, false, b0, (short)0, acc0, false, false);
        acc1 = __builtin_amdgcn_wmma_f32_16x16x4_f32(false, a, false, b1, (short)0, acc1, false, false);
        acc2 = __builtin_amdgcn_wmma_f32_16x16x4_f32(false, a, false, b2, (short)0, acc2, false, false);
        acc3 = __builtin_amdgcn_wmma_f32_16x16x4_f32(false, a, false, b3, (short)0, acc3, false, false);
    }

    float* Yb = Y + tn + l16;
#pragma unroll
    for (int v = 0; v < 8; ++v) {
        int m = tm + v + half * 8;
        long rowoff = (long)m * P;
        float v0 = acc0[v], v1 = acc1[v], v2 = acc2[v], v3 = acc3[v];
        if (bias) {                                  // wave-uniform branch
            float bv = bias[m];
            v0 += bv; v1 += bv; v2 += bv; v3 += bv;
        }
        if (mode == 1) {                             // lr1(0.1) for head_mlp
            v0 = v0 >= 0.f ? v0 : 0.1f * v0;
            v1 = v1 >= 0.f ? v1 : 0.1f * v1;
            v2 = v2 >= 0.f ? v2 : 0.1f * v2;
            v3 = v3 >= 0.f ? v3 : 0.1f * v3;
        }
        Yb[rowoff]      = v0;
        Yb[rowoff + 16] = v1;
        Yb[rowoff + 32] = v2;
        Yb[rowoff + 48] = v3;
    }
}

// 3b. Guarded WMMA GEMM for ragged M (heads: M = 19, 1, 260).
//     mode: 1 lr1(0.1) | 2 lr1+relu | 3 lr1+sigmoid
__global__ void k_gemm_edge(const float* __restrict__ W, const float* __restrict__ X,
                            const float* __restrict__ bias, float* __restrict__ Y,
                            int M, int Kc, long P, int mode) {
    const int wave = threadIdx.x >> 5;
    const int lane = threadIdx.x & 31;
    const long pTiles = P >> 4;
    const int  mTiles = (M + 15) >> 4;
    long tile = (long)blockIdx.x * (blockDim.x >> 5) + wave;
    if (tile >= (long)mTiles * pTiles) return;
    const int  tm = (int)(tile / pTiles) * 16;
    const long tn = (tile % pTiles) * 16;
    const int half  = lane >> 4;
    const int l16   = lane & 15;
    const int kbase = half * 2;
    const int mrow  = tm + l16;
    const bool mok  = (mrow < M);

    v8f acc = {};
    for (int k0 = 0; k0 < Kc; k0 += 4) {            // Kc % 4 == 0 for all users
        int ka = k0 + kbase;
        v2f a, bmat;
        a.x    = mok ? W[(long)mrow * Kc + ka]     : 0.f;
        a.y    = mok ? W[(long)mrow * Kc + ka + 1] : 0.f;
        bmat.x = X[(long)ka       * P + tn + l16];
        bmat.y = X[(long)(ka + 1) * P + tn + l16];
        acc = __builtin_amdgcn_wmma_f32_16x16x4_f32(false, a, false, bmat,
                                                    (short)0, acc, false, false);
    }
#pragma unroll
    for (int v = 0; v < 8; ++v) {
        int m = tm + v + half * 8;
        if (m < M) {
            float val = acc[v];
            if (bias) val += bias[m];
            if (mode >= 1) val = val >= 0.f ? val : 0.1f * val;
            if (mode == 2) val = fmaxf(val, 0.f);
            if (mode == 3) val = 1.f / (1.f + expf(-val));
            Y[(long)m * P + tn + l16] = val;
        }
    }
}

// ---------------------------------------------------------------------------
// 4. BatchNorm stats per channel (block per channel)
__global__ void k_bn_stats(const float* __restrict__ X, long P,
                           float* __restrict__ mean, float* __restrict__ istd) {
    __shared__ float ssum[256], ssq[256];
    int c = blockIdx.x;
    const float* row = X + (long)c * P;
    float s = 0.f, q = 0.f;
    for (long i = threadIdx.x; i < P; i += 256) { float v = row[i]; s += v; q += v * v; }
    ssum[threadIdx.x] = s; ssq[threadIdx.x] = q;
    __syncthreads();
    for (int st = 128; st > 0; st >>= 1) {
        if ((int)threadIdx.x < st) {
            ssum[threadIdx.x] += ssum[threadIdx.x + st];
            ssq[threadIdx.x]  += ssq[threadIdx.x + st];
        }
        __syncthreads();
    }
    if (threadIdx.x == 0) {
        float m = ssum[0] / (float)P;
        float v = ssq[0] / (float)P - m * m;
        mean[c] = m;
        istd[c] = rsqrtf(v + EPSV);
    }
}

// 5. fused BN apply + LeakyReLU(slope)
__global__ void k_bn_lrelu(float* __restrict__ X, const float* __restrict__ mean,
                           const float* __restrict__ istd, const float* __restrict__ g,
                           const float* __restrict__ bb, long P, long total, float slope) {
    long t = (long)blockIdx.x * blockDim.x + threadIdx.x;
    if (t >= total) return;
    int c = (int)(t / P);
    float v = (X[t] - mean[c]) * istd[c] * g[c] + bb[c];
    X[t] = v >= 0.f ? v : slope * v;
}

// ---------------------------------------------------------------------------
// 6. max over K: X[C][BN*K] -> Y[C][BN]   (contiguous K)
__global__ void k_maxk(const float* __restrict__ X, float* __restrict__ Y, long total) {
    long t = (long)blockIdx.x * blockDim.x + threadIdx.x;
    if (t >= total) return;
    const float* src = X + t * Kk;
    float m = src[0];
#pragma unroll
    for (int k = 1; k < Kk; ++k) m = fmaxf(m, src[k]);
    Y[t] = m;
}

// 7. max over N per (b,c): X[C][BN] -> Y[b*C + c]
__global__ void k_maxn(const float* __restrict__ X, float* __restrict__ Y, int C) {
    int t = blockIdx.x * blockDim.x + threadIdx.x;
    if (t >= Bb * C) return;
    int b = t / C, c = t % C;
    const float* row = X + (long)c * BN + (long)b * Nn;
    float m = row[0];
    for (int n = 1; n < Nn; ++n) m = fmaxf(m, row[n]);
    Y[t] = m;
}

// ---------------------------------------------------------------------------
// 8. T-Net tail: lin1+bn2d+lrelu, lin2+bn2d+lrelu, 3x3 transform -> t4m[2][16]
__global__ void k_tnet_tail(const float* __restrict__ h512,
                            const float* __restrict__ lin1_w, const float* __restrict__ g1, const float* __restrict__ b1,
                            const float* __restrict__ lin2_w, const float* __restrict__ g2, const float* __restrict__ b2,
                            const float* __restrict__ trans_w, const float* __restrict__ trans_b,
                            float* __restrict__ t4m) {
    __shared__ float s1[2 * 256];
    __shared__ float s2[2 * 128];
    int t = threadIdx.x;
    if (t < 256) {
        for (int b = 0; b < 2; ++b) {
            float acc = 0.f;
            for (int c = 0; c < 512; ++c) acc += h512[b * 512 + c] * lin1_w[t * 512 + c];
            s1[b * 256 + t] = acc;
        }
    }
    __syncthreads();
    if (t < 256) {
        float a = s1[t], c2 = s1[256 + t];
        float m = 0.5f * (a + c2);
        float v = 0.5f * ((a - m) * (a - m) + (c2 - m) * (c2 - m));
        float is = rsqrtf(v + EPSV);
        float ya = (a  - m) * is * g1[t] + b1[t]; ya = ya >= 0.f ? ya : 0.2f * ya;
        float yb = (c2 - m) * is * g1[t] + b1[t]; yb = yb >= 0.f ? yb : 0.2f * yb;
        s1[t] = ya; s1[256 + t] = yb;
    }
    __syncthreads();
    if (t < 128) {
        for (int b = 0; b < 2; ++b) {
            float acc = 0.f;
            for (int c = 0; c < 256; ++c) acc += s1[b * 256 + c] * lin2_w[t * 256 + c];
            s2[b * 128 + t] = acc;
        }
    }
    __syncthreads();
    if (t < 128) {
        float a = s2[t], c2 = s2[128 + t];
        float m = 0.5f * (a + c2);
        float v = 0.5f * ((a - m) * (a - m) + (c2 - m) * (c2 - m));
        float is = rsqrtf(v + EPSV);
        float ya = (a  - m) * is * g2[t] + b2[t]; ya = ya >= 0.f ? ya : 0.2f * ya;
        float yb = (c2 - m) * is * g2[t] + b2[t]; yb = yb >= 0.f ? yb : 0.2f * yb;
        s2[t] = ya; s2[128 + t] = yb;
    }
    __syncthreads();
    if (t < 32) {                                // identity part of 4x4
        int e = t % 16, cc = e / 4, dd = e % 4;
        if (!(cc < 3 && dd < 3)) t4m[t] = (cc == 3 && dd == 3) ? 1.f : 0.f;
    }
    if (t < 18) {                                // 3x3 block
        int b = t / 9, j = t % 9;
        float acc = trans_b[j];
        for (int c = 0; c < 128; ++c) acc += s2[b * 128 + c] * trans_w[j * 128 + c];
        t4m[b * 16 + (j / 3) * 4 + (j % 3)] = acc;
    }
}

// 9. apply 4x4 transform per point
__global__ void k_transform(const float* __restrict__ xc0, const float* __restrict__ t4m,
                            float* __restrict__ xc1) {
    int p = blockIdx.x * blockDim.x + threadIdx.x;
    if (p >= BN) return;
    int b = p / Nn;
    float v0 = xc0[p], v1 = xc0[BN + p], v2 = xc0[2 * BN + p], v3 = xc0[3 * BN + p];
    const float* T = t4m + b * 16;
#pragma unroll
    for (int d = 0; d < 4; ++d)
        xc1[(long)d * BN + p] = v0 * T[d] + v1 * T[4 + d] + v2 * T[8 + d] + v3 * T[12 + d];
}

// ---------------------------------------------------------------------------
// 10. utility kernels
__global__ void k_copy(float* __restrict__ dst, const float* __restrict__ src, long n) {
    long t = (long)blockIdx.x * blockDim.x + threadIdx.x;
    if (t < n) dst[t] = src[t];
}
__global__ void k_zero(float* __restrict__ dst, long n) {
    long t = (long)blockIdx.x * blockDim.x + threadIdx.x;
    if (t < n) dst[t] = 0.f;
}
__global__ void k_padw(const float* __restrict__ W, float* __restrict__ Wp,
                       int M, int Kc, int Kp) {
    int t = blockIdx.x * blockDim.x + threadIdx.x;
    if (t >= M * Kp) return;
    int m = t / Kp, k = t % Kp;
    Wp[t] = (k < Kc) ? W[m * Kc + k] : 0.f;
}
__global__ void k_bcast(float* __restrict__ dst, const float* __restrict__ emax, int C) {
    long t = (long)blockIdx.x * blockDim.x + threadIdx.x;    // over C*BN
    if (t >= (long)C * BN) return;
    int c = (int)(t / BN);
    int p = (int)(t % BN);
    int b = p / Nn;
    dst[t] = emax[b * C + c];
}
__global__ void k_emit(float* __restrict__ dst, const float* __restrict__ src, int C) {
    long t = (long)blockIdx.x * blockDim.x + threadIdx.x;    // over C*BN
    if (t >= (long)C * BN) return;
    int c = (int)(t / BN);
    int p = (int)(t % BN);
    dst[(long)p * C + c] = src[(long)c * BN + p];
}

// ---------------------------------------------------------------------------
static void run_gemm(const float* W, const float* X, float* Y,
                     int M, int Kc, long P, const float* bias, int mode, hipStream_t s) {
    long waves = (long)(M >> 4) * (P >> 6);
    int blocks = (int)((waves + 7) / 8);                     // 8 waves / block
    hipLaunchKernelGGL(k_gemm_fast, dim3(blocks), dim3(256), 0, s, W, X, Y, M, Kc, P, bias, mode);
}
static void run_gemm_edge(const float* W, const float* X, const float* bias, float* Y,
                          int M, int Kc, long P, int mode, hipStream_t s) {
    long tiles = (long)((M + 15) / 16) * (P / 16);
    int blocks = (int)((tiles + 7) / 8);
    hipLaunchKernelGGL(k_gemm_edge, dim3(blocks), dim3(256), 0, s, W, X, bias, Y, M, Kc, P, mode);
}
static void run_bn(float* X, int C, long P, const float* g, const float* bb,
                   float* mean, float* istd, float slope, hipStream_t s) {
    hipLaunchKernelGGL(k_bn_stats, dim3(C), dim3(256), 0, s, X, P, mean, istd);
    long total = (long)C * P;
    hipLaunchKernelGGL(k_bn_lrelu, dim3((int)((total + 255) / 256)), dim3(256), 0, s,
                       X, mean, istd, g, bb, P, total, slope);
}

extern "C" void kernel_launch(void* const* d_in, const int* in_sizes, int n_in,
                              void* d_out, int out_size, void* d_ws, size_t ws_size,
                              hipStream_t stream) {
    const float* x = (const float*)d_in[0];
    #define P_(i) ((const float*)d_in[i])
    const float *t_conv1_w = P_(1),  *t_bn1_g = P_(2),  *t_bn1_b = P_(3);
    const float *t_conv2_w = P_(4),  *t_bn2_g = P_(5),  *t_bn2_b = P_(6);
    const float *t_conv3_w = P_(7),  *t_bn3a_g = P_(8), *t_bn3a_b = P_(9);
    const float *t_lin1_w = P_(10),  *t_bn3b_g = P_(11), *t_bn3b_b = P_(12);
    const float *t_lin2_w = P_(13),  *t_bn4g = P_(14),  *t_bn4b = P_(15);
    const float *t_trans_w = P_(16), *t_trans_b = P_(17);
    const float *conv1_w = P_(18), *bn1_g = P_(19), *bn1_b = P_(20);
    const float *conv2_w = P_(21), *bn2_g = P_(22), *bn2_b = P_(23);
    const float *conv3_w = P_(24), *bn3_g = P_(25), *bn3_b = P_(26);
    const float *conv4_w = P_(27), *bn4_g = P_(28), *bn4_b = P_(29);
    const float *conv5_w = P_(30), *bn5_g = P_(31), *bn5_b = P_(32);
    const float *conv6_w = P_(33), *bn6_g = P_(34), *bn6_b = P_(35);
    const float *conv7_w = P_(36), *bn7_g = P_(37), *bn7_b = P_(38);
    const float *conv8_w = P_(39), *bn8_g = P_(40), *bn8_b = P_(41);
    const float *hm_w = P_(42), *hm_b = P_(43);
    const float *hc_w = P_(44), *hc_b = P_(45);
    const float *hv_w = P_(46), *hv_b = P_(47);
    const float *hs_w = P_(48), *hs_b = P_(49);
    #undef P_

    // workspace layout (floats)
    float* ws = (float*)d_ws;
    size_t off = 0;
    auto A = [&](size_t n) { float* p = ws + off; off += n; return p; };
    float* xc0     = A((size_t)4 * BN);
    float* xc1     = A((size_t)4 * BN);
    int*   idx     = (int*)A((size_t)BNK);
    float* E       = A((size_t)128 * BNK);       // edge / ping
    float* Y1      = A((size_t)64 * BNK);        // pong
    float* t128    = A((size_t)128 * BN);
    float* t512    = A((size_t)512 * BN);
    float* h512    = A((size_t)2 * 512);
    float* t4m     = A(32);
    float* x1      = A((size_t)64 * BN);
    float* x2      = A((size_t)64 * BN);
    float* x3      = A((size_t)64 * BN);
    float* cat192  = A((size_t)192 * BN);
    float* e1024   = A((size_t)1024 * BN);
    float* emax    = A((size_t)2 * 1024);
    float* cat1216 = A((size_t)1216 * BN);
    float* c512    = A((size_t)512 * BN);
    float* c256    = A((size_t)256 * BN);
    float* f256    = A((size_t)256 * BN);
    float* o19     = A((size_t)19 * BN);
    float* c1      = A((size_t)BN);
    float* v260    = A((size_t)260 * BN);
    float* meanb   = A(2048);
    float* istdb   = A(2048);
    float* w1pad   = A(64 * 8);                  // t_conv1 weight padded 6->8

    const int BNblk  = BN / 256;
    const int BNKblk = BNK / 256;

    hipLaunchKernelGGL(k_transpose_in, dim3(BNblk), dim3(256), 0, stream, x, xc0);
    hipLaunchKernelGGL(k_padw, dim3(2), dim3(256), 0, stream, t_conv1_w, w1pad, 64, 6, 8);

    // ---- T-Net branch ----
    hipLaunchKernelGGL(k_knn, dim3(BN / 128), dim3(128), 0, stream, xc0, idx);
    hipLaunchKernelGGL(k_edge, dim3(BNKblk), dim3(256), 0, stream, xc0, idx, E, 3);
    hipLaunchKernelGGL(k_zero, dim3((int)(((long)2 * BNK + 255) / 256)), dim3(256), 0, stream,
                       E + (long)6 * BNK, (long)2 * BNK);    // pad rows 6,7
    run_gemm(w1pad, E, Y1, 64, 8, BNK, nullptr, 0, stream);
    run_bn(Y1, 64, BNK, t_bn1_g, t_bn1_b, meanb, istdb, 0.2f, stream);
    run_gemm(t_conv2_w, Y1, E, 128, 64, BNK, nullptr, 0, stream);
    run_bn(E, 128, BNK, t_bn2_g, t_bn2_b, meanb, istdb, 0.2f, stream);
    hipLaunchKernelGGL(k_maxk, dim3((128 * BN) / 256), dim3(256), 0, stream, E, t128, (long)128 * BN);
    run_gemm(t_conv3_w, t128, t512, 512, 128, BN, nullptr, 0, stream);
    run_bn(t512, 512, BN, t_bn3a_g, t_bn3a_b, meanb, istdb, 0.2f, stream);
    hipLaunchKernelGGL(k_maxn, dim3((2 * 512 + 255) / 256), dim3(256), 0, stream, t512, h512, 512);
    hipLaunchKernelGGL(k_tnet_tail, dim3(1), dim3(256), 0, stream, h512,
                       t_lin1_w, t_bn3b_g, t_bn3b_b, t_lin2_w, t_bn4g, t_bn4b,
                       t_trans_w, t_trans_b, t4m);
    hipLaunchKernelGGL(k_transform, dim3(BNblk), dim3(256), 0, stream, xc0, t4m, xc1);

    // ---- EdgeConv stack ----
    hipLaunchKernelGGL(k_knn, dim3(BN / 128), dim3(128), 0, stream, xc1, idx);
    hipLaunchKernelGGL(k_edge, dim3(BNKblk), dim3(256), 0, stream, xc1, idx, E, 4);
    run_gemm(conv1_w, E, Y1, 64, 8, BNK, nullptr, 0, stream);
    run_bn(Y1, 64, BNK, bn1_g, bn1_b, meanb, istdb, 0.2f, stream);
    run_gemm(conv2_w, Y1, E, 64, 64, BNK, nullptr, 0, stream);
    run_bn(E, 64, BNK, bn2_g, bn2_b, meanb, istdb, 0.2f, stream);
    hipLaunchKernelGGL(k_maxk, dim3((64 * BN) / 256), dim3(256), 0, stream, E, x1, (long)64 * BN);

    hipLaunchKernelGGL(k_knn, dim3(BN / 128), dim3(128), 0, stream, x1, idx);
    hipLaunchKernelGGL(k_edge, dim3(BNKblk), dim3(256), 0, stream, x1, idx, E, 64);
    run_gemm(conv3_w, E, Y1, 64, 128, BNK, nullptr, 0, stream);
    run_bn(Y1, 64, BNK, bn3_g, bn3_b, meanb, istdb, 0.2f, stream);
    run_gemm(conv4_w, Y1, E, 64, 64, BNK, nullptr, 0, stream);
    run_bn(E, 64, BNK, bn4_g, bn4_b, meanb, istdb, 0.2f, stream);
    hipLaunchKernelGGL(k_maxk, dim3((64 * BN) / 256), dim3(256), 0, stream, E, x2, (long)64 * BN);

    hipLaunchKernelGGL(k_knn, dim3(BN / 128), dim3(128), 0, stream, x2, idx);
    hipLaunchKernelGGL(k_edge, dim3(BNKblk), dim3(256), 0, stream, x2, idx, E, 64);
    run_gemm(conv5_w, E, Y1, 64, 128, BNK, nullptr, 0, stream);
    run_bn(Y1, 64, BNK, bn5_g, bn5_b, meanb, istdb, 0.2f, stream);
    hipLaunchKernelGGL(k_maxk, dim3((64 * BN) / 256), dim3(256), 0, stream, Y1, x3, (long)64 * BN);

    // ---- global feature + decoder ----
    long c64 = (long)64 * BN;
    int  cb  = (int)((c64 + 255) / 256);
    hipLaunchKernelGGL(k_copy, dim3(cb), dim3(256), 0, stream, cat192,            x1, c64);
    hipLaunchKernelGGL(k_copy, dim3(cb), dim3(256), 0, stream, cat192 + 64 * BN,  x2, c64);
    hipLaunchKernelGGL(k_copy, dim3(cb), dim3(256), 0, stream, cat192 + 128 * BN, x3, c64);
    run_gemm(conv6_w, cat192, e1024, 1024, 192, BN, nullptr, 0, stream);
    run_bn(e1024, 1024, BN, bn6_g, bn6_b, meanb, istdb, 0.2f, stream);
    hipLaunchKernelGGL(k_maxn, dim3((2 * 1024 + 255) / 256), dim3(256), 0, stream, e1024, emax, 1024);
    long c1024 = (long)1024 * BN;
    hipLaunchKernelGGL(k_bcast, dim3((int)((c1024 + 255) / 256)), dim3(256), 0, stream, cat1216, emax, 1024);
    hipLaunchKernelGGL(k_copy, dim3(cb), dim3(256), 0, stream, cat1216 + (long)1024 * BN, x1, c64);
    hipLaunchKernelGGL(k_copy, dim3(cb), dim3(256), 0, stream, cat1216 + (long)1088 * BN, x2, c64);
    hipLaunchKernelGGL(k_copy, dim3(cb), dim3(256), 0, stream, cat1216 + (long)1152 * BN, x3, c64);
    run_gemm(conv7_w, cat1216, c512, 512, 1216, BN, nullptr, 0, stream);
    run_bn(c512, 512, BN, bn7_g, bn7_b, meanb, istdb, 0.2f, stream);
    run_gemm(conv8_w, c512, c256, 256, 512, BN, nullptr, 0, stream);
    run_bn(c256, 256, BN, bn8_g, bn8_b, meanb, istdb, 0.2f, stream);

    // ---- heads ----
    run_gemm(hm_w, c256, f256, 256, 256, BN, hm_b, 1, stream);        // f = lr1(.)
    run_gemm_edge(hs_w, f256, hs_b, o19, 19, 256, BN, 1, stream);     // out = lr1
    run_gemm_edge(hc_w, f256, hc_b, c1, 1, 256, BN, 3, stream);       // c = sigmoid(lr1)
    run_gemm_edge(hv_w, f256, hv_b, v260, 260, 256, BN, 2, stream);   // v = relu(lr1)

    // ---- emit (B,N,C) concatenated: out, c, v, f ----
    float* dout = (float*)d_out;
    hipLaunchKernelGGL(k_emit, dim3((int)(((long)19 * BN + 255) / 256)), dim3(256), 0, stream,
                       dout, o19, 19);
    hipLaunchKernelGGL(k_emit, dim3((BN + 255) / 256), dim3(256), 0, stream,
                       dout + (long)19 * BN, c1, 1);
    hipLaunchKernelGGL(k_emit, dim3((int)(((long)260 * BN + 255) / 256)), dim3(256), 0, stream,
                       dout + (long)20 * BN, v260, 260);
    hipLaunchKernelGGL(k_emit, dim3((int)(((long)256 * BN + 255) / 256)), dim3(256), 0, stream,
                       dout + (long)280 * BN, f256, 256);
}